// SpikingCriticTwin_21912923144775
// MI455X (gfx1250) — compile-verified
//
#include <hip/hip_runtime.h>

typedef __attribute__((ext_vector_type(2))) float v2f;
typedef __attribute__((ext_vector_type(8))) float v8f;

#define BM      32        // batch rows per block
#define HDIM    512
#define DDIM    393       // 376 state + 17 action
#define DPAD    404       // phase-1 K padded to multiple of 4
#define XSTR    404       // x tile row stride (floats)  -> conflict-free A b64 reads
#define S1STR   516       // s1 row stride (floats)      -> conflict-free A b64 reads
#define RP      1056      // packed K-pair pitch (floats); 1056 % 64 == 32 -> conflict-free B b64
#define KS      16        // K rows per slab (= 8 packed pairs)
#define NPAIR   (KS/2)
#define SLABF   (NPAIR*RP)   // 8448 floats per slab buffer
#define TSTEPS  8

// LDS partition (float offsets)
#define OFF_H   0
#define OFF_V1  (BM*HDIM)                 // 16384
#define OFF_S1  (2*BM*HDIM)               // 32768 (x tile in phase 1)
#define OFF_W   (OFF_S1 + BM*S1STR)       // 49280 : two slab buffers
#define OFF_DV  (OFF_W + 2*SLABF)         // 66176
#define OFF_V3  (OFF_DV + 32)
#define LDS_FLOATS (OFF_V3 + 32)          // 66240 floats = 264960 B (< 320KB WGP LDS)

__global__ __launch_bounds__(256)
void snn_twin_critic_kernel(
    const float* __restrict__ state, const float* __restrict__ action,
    const float* __restrict__ W1a, const float* __restrict__ b1a,
    const float* __restrict__ W2a, const float* __restrict__ b2a,
    const float* __restrict__ W3a, const float* __restrict__ b3a,
    const float* __restrict__ W1b, const float* __restrict__ b1b,
    const float* __restrict__ W2b, const float* __restrict__ b2b,
    const float* __restrict__ W3b, const float* __restrict__ b3b,
    float* __restrict__ out, int batch)
{
    extern __shared__ __align__(16) float lds[];
    float* hS  = lds + OFF_H;
    float* v1S = lds + OFF_V1;
    float* s1S = lds + OFF_S1;
    float* wB0 = lds + OFF_W;
    float* wB1 = lds + OFF_W + SLABF;
    float* dvS = lds + OFF_DV;
    float* v3S = lds + OFF_V3;

    const int tid  = threadIdx.x;
    const int lane = tid & 31;
    const int wave = tid >> 5;
    const int l16  = lane & 15;
    const int half = lane >> 4;            // 0: lanes 0-15, 1: lanes 16-31
    const int wm   = wave >> 2;            // 0..1 -> 16-row half of the tile
    const int wn   = wave & 3;             // 0..3 -> 128-col chunk
    const int r0   = blockIdx.x * BM;
    const int br   = blockIdx.y;           // branch (Q1 / Q2)

    const float* W1 = br ? W1b : W1a;  const float* B1 = br ? b1b : b1a;
    const float* W2 = br ? W2b : W2a;  const float* B2 = br ? b2b : b2a;
    const float* W3 = br ? W3b : W3a;  const float* B3 = br ? b3b : b3a;

    // Per-lane column constants for the 8 N-tiles this wave owns.
    float b1c[8], b2c[8], w3c[8];
    #pragma unroll
    for (int t = 0; t < 8; ++t) {
        int col = wn*128 + t*16 + l16;
        b1c[t] = B1[col]; b2c[t] = B2[col]; w3c[t] = W3[col];
    }

    const int aRow = wm*16 + l16;          // A-fragment row for this lane

    // -------- slab staging: wave stages K-pair 'wave', lane covers 16 cols --------
    float4 sa[4], sb[4];                   // register staging: rows 2p and 2p+1
    const int cb0 = lane * 16;

    auto loadSlab = [&](const float* __restrict__ W, int ks, int klimit) {
        int rA = ks + 2*wave, rB = rA + 1;
        #pragma unroll
        for (int q = 0; q < 4; ++q) {
            sa[q] = (rA < klimit) ? *(const float4*)&W[rA*HDIM + cb0 + 4*q]
                                  : make_float4(0.f, 0.f, 0.f, 0.f);
            sb[q] = (rB < klimit) ? *(const float4*)&W[rB*HDIM + cb0 + 4*q]
                                  : make_float4(0.f, 0.f, 0.f, 0.f);
        }
    };
    auto storeSlab = [&](float* buf) {     // pair-interleaved: [pair][col][k&1]
        float* dst = buf + wave*RP + 2*cb0;
        #pragma unroll
        for (int q = 0; q < 4; ++q) {
            *(float4*)&dst[8*q]     = make_float4(sa[q].x, sb[q].x, sa[q].y, sb[q].y);
            *(float4*)&dst[8*q + 4] = make_float4(sa[q].z, sb[q].z, sa[q].w, sb[q].w);
        }
    };

    v8f acc[8];
    #pragma unroll
    for (int t = 0; t < 8; ++t)
        #pragma unroll
        for (int i = 0; i < 8; ++i) acc[t][i] = 0.0f;

    auto computeSlab = [&](const float* buf, const float* aBase, int astr,
                           int k0base, int kmax) {
        #pragma unroll
        for (int kk = 0; kk < KS; kk += 4) {
            if (kk >= kmax) break;                       // wave-uniform
            v2f a = *(const v2f*)&aBase[aRow*astr + k0base + kk + 2*half];
            const float* bp = buf + (kk/2 + half)*RP + 2*(wn*128 + l16);
            #pragma unroll
            for (int t = 0; t < 8; ++t) {
                v2f b = *(const v2f*)&bp[32*t];          // single b64, conflict-free
                acc[t] = __builtin_amdgcn_wmma_f32_16x16x4_f32(
                    false, a, false, b, (short)0, acc[t], false, false);
            }
        }
    };

    // ---------------- Phase 1: h = x @ W1 + b1 ----------------
    for (int idx = tid; idx < BM*XSTR; idx += 256) {     // x tile (zero-padded)
        int r = idx / XSTR, c = idx - r*XSTR;
        float v = 0.0f;
        if (c < 376)       v = state[(r0 + r)*376 + c];
        else if (c < DDIM) v = action[(r0 + r)*17 + (c - 376)];
        s1S[r*XSTR + c] = v;
    }
    const int nslab1 = (DPAD + KS - 1) / KS;             // 26
    loadSlab(W1, 0, DDIM);
    storeSlab(wB0);
    __syncthreads();
    for (int s = 0; s < nslab1; ++s) {
        if (s + 1 < nslab1) loadSlab(W1, (s+1)*KS, DDIM);
        int kmax = DPAD - s*KS; if (kmax > KS) kmax = KS;
        computeSlab((s & 1) ? wB1 : wB0, s1S, XSTR, s*KS, kmax);
        if (s + 1 < nslab1) storeSlab((s & 1) ? wB0 : wB1);
        __syncthreads();
    }

    // h -> LDS (C-layout rows), init v1 / v2 / v3 / dv.
    #pragma unroll
    for (int t = 0; t < 8; ++t) {
        #pragma unroll
        for (int i = 0; i < 8; ++i) {
            int row = wm*16 + i + 8*half;
            int col = wn*128 + t*16 + l16;
            hS[row*HDIM + col] = acc[t][i] + b1c[t];
            acc[t][i] = 0.0f;                            // acc becomes persistent v2
        }
    }
    for (int q = tid; q < BM*HDIM/4; q += 256)
        *(float4*)&v1S[4*q] = make_float4(0.f, 0.f, 0.f, 0.f);
    if (tid < 32) { dvS[tid] = 0.0f; v3S[tid] = 0.0f; }
    __syncthreads();

    // ---------------- Phase 2: 8 fused timesteps ----------------
    for (int step = 0; step < TSTEPS; ++step) {
        // (a) IF neuron 1: charge, fire, hard-reset; emit s1 (0/1 as f32)
        for (int q = tid; q < BM*HDIM/4; q += 256) {
            int idx = 4*q;
            float4 v = *(float4*)&v1S[idx];
            float4 h = *(float4*)&hS[idx];
            float4 s, vn;
            v.x += h.x; s.x = (v.x >= 1.f) ? 1.f : 0.f; vn.x = (v.x >= 1.f) ? 0.f : v.x;
            v.y += h.y; s.y = (v.y >= 1.f) ? 1.f : 0.f; vn.y = (v.y >= 1.f) ? 0.f : v.y;
            v.z += h.z; s.z = (v.z >= 1.f) ? 1.f : 0.f; vn.z = (v.z >= 1.f) ? 0.f : v.z;
            v.w += h.w; s.w = (v.w >= 1.f) ? 1.f : 0.f; vn.w = (v.w >= 1.f) ? 0.f : v.w;
            int r = idx >> 9, c = idx & 511;
            *(float4*)&s1S[r*S1STR + c] = s;
            *(float4*)&v1S[idx] = vn;
        }
        // (b) v2 += s1 @ W2 : exact f32 WMMA, double-buffered K-slabs
        loadSlab(W2, 0, HDIM);
        storeSlab(wB0);
        __syncthreads();                                  // covers (a) and slab 0
        for (int s = 0; s < HDIM/KS; ++s) {               // 32 slabs
            if (s + 2 < HDIM/KS)                          // keep L2 warm on first pass
                __builtin_prefetch(&W2[(s+2)*KS*HDIM] + tid*32, 0, 1);
            if (s + 1 < HDIM/KS) loadSlab(W2, (s+1)*KS, HDIM);
            computeSlab((s & 1) ? wB1 : wB0, s1S, S1STR, s*KS, KS);
            if (s + 1 < HDIM/KS) storeSlab((s & 1) ? wB0 : wB1);
            __syncthreads();
        }
        // (c) IF neuron 2: +b2, fire, reset; dv partial = s2 . W3
        float p[8];
        #pragma unroll
        for (int i = 0; i < 8; ++i) p[i] = 0.0f;
        #pragma unroll
        for (int t = 0; t < 8; ++t) {
            #pragma unroll
            for (int i = 0; i < 8; ++i) {
                float a2 = acc[t][i] + b2c[t];
                bool fire = (a2 >= 1.0f);
                p[i] += fire ? w3c[t] : 0.0f;
                acc[t][i] = fire ? 0.0f : a2;             // v2 persists in accumulators
            }
        }
        #pragma unroll
        for (int i = 0; i < 8; ++i) {                     // reduce 16 lanes of a row
            p[i] += __shfl_xor(p[i], 1, 32);
            p[i] += __shfl_xor(p[i], 2, 32);
            p[i] += __shfl_xor(p[i], 4, 32);
            p[i] += __shfl_xor(p[i], 8, 32);
        }
        if (l16 == 0) {
            #pragma unroll
            for (int i = 0; i < 8; ++i)
                atomicAdd(&dvS[wm*16 + i + 8*half], p[i]);
        }
        __syncthreads();
        // (d) non-spiking LIF output: v3 += (dv - v3)/tau
        if (tid < 32) {
            float d   = dvS[tid] + B3[0];
            float v3o = v3S[tid];
            v3S[tid]  = v3o + (d - v3o) * 0.5f;
            dvS[tid]  = 0.0f;
        }
        __syncthreads();
    }

    if (tid < 32) out[br*batch + r0 + tid] = v3S[tid];
}

extern "C" void kernel_launch(void* const* d_in, const int* in_sizes, int n_in,
                              void* d_out, int out_size, void* d_ws, size_t ws_size,
                              hipStream_t stream) {
    (void)n_in; (void)out_size; (void)d_ws; (void)ws_size;
    const float* state  = (const float*)d_in[0];
    const float* action = (const float*)d_in[1];
    const float* W1a = (const float*)d_in[2];  const float* b1a = (const float*)d_in[3];
    const float* W2a = (const float*)d_in[4];  const float* b2a = (const float*)d_in[5];
    const float* W3a = (const float*)d_in[6];  const float* b3a = (const float*)d_in[7];
    const float* W1b = (const float*)d_in[8];  const float* b1b = (const float*)d_in[9];
    const float* W2b = (const float*)d_in[10]; const float* b2b = (const float*)d_in[11];
    const float* W3b = (const float*)d_in[12]; const float* b3b = (const float*)d_in[13];
    float* out = (float*)d_out;

    const int batch = in_sizes[0] / 376;          // 4096
    dim3 grid(batch / BM, 2), block(256);
    size_t ldsBytes = (size_t)LDS_FLOATS * sizeof(float);
    snn_twin_critic_kernel<<<grid, block, ldsBytes, stream>>>(
        state, action, W1a, b1a, W2a, b2a, W3a, b3a,
        W1b, b1b, W2b, b2b, W3b, b3b, out, batch);
}